// E3PhisNetHamiltonian_29609504538689
// MI455X (gfx1250) — compile-verified
//
#include <hip/hip_runtime.h>

// ============================================================================
// E3 PhisNet Hamiltonian head for MI455X (gfx1250, wave32).
// f32 end-to-end (workload is tiny & latency-bound on a 23.3 TB/s part).
// WMMA (V_WMMA_F32_16X16X4_F32) used for the three GEMM-shaped stages:
//   - per-irrep channel mixing (16x16 weights over channel axis)
//   - pairmix edge-weight GEMM (16384x32 @ 32x240)
//   - per-irrep output linear (mul_out x 16 over channel axis)
// CG tensor contractions are VALU; CG table built on-device per launch.
// CG table offsets are evaluated at compile time into __constant__ memory.
// ============================================================================

typedef float v2f __attribute__((ext_vector_type(2)));
typedef float v8f __attribute__((ext_vector_type(8)));

#define DEVI __device__ __forceinline__

DEVI int iabs_(int a){ return a < 0 ? -a : a; }

// Flattened CG table offset; iteration order fixed (l1 outer, l2, l3 inner).
__host__ __device__ constexpr int cg_off_ce(int L1, int L2, int L3) {
  int off = 0;
  for (int a = 0; a <= 4; ++a)
    for (int b = 0; b <= 4; ++b) {
      int lo = (a > b) ? a - b : b - a;
      int hi = (a + b < 4) ? a + b : 4;
      for (int c = lo; c <= hi; ++c) {
        if (a == L1 && b == L2 && c == L3) return off;
        off += (2*a+1)*(2*b+1)*(2*c+1);
      }
    }
  return 0;
}
// total = 13075 floats

struct CgTab { int v[125]; };
constexpr CgTab make_cgtab() {
  CgTab t{};
  for (int i = 0; i < 125; ++i) t.v[i] = cg_off_ce(i/25, (i/5)%5, i%5);
  return t;
}
__constant__ CgTab CGOFF = make_cgtab();
DEVI int cg_off(int l1, int l2, int l3) { return CGOFF.v[l1*25 + l2*5 + l3]; }

// ---------------------------------------------------------------------------
// CG table construction (device, double precision)
// ---------------------------------------------------------------------------
__device__ inline double dev_su2_cg(int j1,int m1,int j2,int m2,int j3,int m3){
  if (m1 + m2 != m3) return 0.0;
  double f[16];
  f[0] = 1.0;
  for (int i = 1; i < 16; ++i) f[i] = f[i-1] * (double)i;
  int vmin = -j1 + j2 + m3;
  if (-j1 + m1 > vmin) vmin = -j1 + m1;
  if (0 > vmin) vmin = 0;
  int vmax = j2 + j3 + m1;
  if (j3 - j1 + j2 < vmax) vmax = j3 - j1 + j2;
  if (j3 + m3 < vmax) vmax = j3 + m3;
  double C = sqrt((2.0*j3 + 1.0) * f[j3+j1-j2] * f[j3-j1+j2] * f[j1+j2-j3]
                  * f[j3+m3] * f[j3-m3]
                  / (f[j1+j2+j3+1] * f[j1-m1] * f[j1+m1] * f[j2-m2] * f[j2+m2]));
  double S = 0.0;
  for (int v = vmin; v <= vmax; ++v) {
    double sg = ((v + j2 + m2) & 1) ? -1.0 : 1.0;
    S += sg * f[j2+j3+m1-v] * f[j1-m1+v]
         / (f[v] * f[j3-j1+j2-v] * f[j3+m3-v] * f[v+j1-j2-m3]);
  }
  return C * S;
}

// Q(l)[row r (complex m), col c (real)] including (-i)^l factor.
__device__ inline void q_entry(int l, int r, int c, double* re, double* im){
  const double is2 = 0.70710678118654752440;
  int mr = r - l;
  double ar = 0.0, ai = 0.0;
  if (mr < 0) {
    if (c == 2*l - r)      ar = is2;       // 1/sqrt2
    else if (c == r)       ai = -is2;      // -i/sqrt2
  } else if (mr == 0) {
    if (c == l) ar = 1.0;
  } else {
    double s = (mr & 1) ? -1.0 : 1.0;      // (-1)^m
    if (c == r)            ar = s * is2;
    else if (c == 2*l - r) ai = s * is2;
  }
  // multiply by (-i)^l
  double br, bi;
  switch (l & 3) {
    case 0: br = ar;  bi = ai;  break;
    case 1: br = ai;  bi = -ar; break;
    case 2: br = -ar; bi = -ai; break;
    default: br = -ai; bi = ar; break;
  }
  *re = br; *im = bi;
}

__global__ void cg_init_kernel(float* __restrict__ cg){
  int target = blockIdx.x;
  int l1 = 0, l2 = 0, l3 = 0; long off = 0; int cnt = 0; bool found = false;
  for (int a = 0; a <= 4 && !found; ++a)
    for (int b = 0; b <= 4 && !found; ++b) {
      int lo = (a > b) ? a - b : b - a;
      int hi = (a + b < 4) ? a + b : 4;
      for (int c = lo; c <= hi; ++c) {
        if (cnt == target) { l1 = a; l2 = b; l3 = c; found = true; break; }
        cnt++;
        off += (long)(2*a+1)*(2*b+1)*(2*c+1);
      }
    }
  if (!found) return;
  int d2 = 2*l2+1, d3 = 2*l3+1;
  int tot = (2*l1+1) * d2 * d3;
  double inv = 1.0 / sqrt((double)(2*l3 + 1));   // e3nn normalization
  for (int t = threadIdx.x; t < tot; t += blockDim.x) {
    int a3 = t % d3; int r = t / d3; int a2 = r % d2; int a1 = r / d2;
    int iset[2]; int ni = 1; iset[0] = a1; if (2*l1 - a1 != a1) { iset[1] = 2*l1 - a1; ni = 2; }
    int kset[2]; int nk = 1; kset[0] = a2; if (2*l2 - a2 != a2) { kset[1] = 2*l2 - a2; nk = 2; }
    int nset[2]; int nn = 1; nset[0] = a3; if (2*l3 - a3 != a3) { nset[1] = 2*l3 - a3; nn = 2; }
    double acc = 0.0;
    for (int ii = 0; ii < ni; ++ii)
      for (int kk = 0; kk < nk; ++kk)
        for (int zz = 0; zz < nn; ++zz) {
          int i = iset[ii], k = kset[kk], n = nset[zz];
          int m1 = i - l1, m2 = k - l2, m3 = n - l3;
          if (m1 + m2 != m3) continue;
          double cval = dev_su2_cg(l1, m1, l2, m2, l3, m3) * inv;
          if (cval == 0.0) continue;
          double q1r,q1i,q2r,q2i,q3r,q3i;
          q_entry(l1, i, a1, &q1r, &q1i);
          q_entry(l2, k, a2, &q2r, &q2i);
          q_entry(l3, n, a3, &q3r, &q3i);
          double tr = q1r*q2r - q1i*q2i, ti = q1r*q2i + q1i*q2r;
          double ur = tr*q3r + ti*q3i;  // Re( q1*q2*conj(q3) )
          acc += ur * cval;
        }
    cg[off + t] = (float)acc;
  }
}

// ---------------------------------------------------------------------------
// Parameter bundles (passed by value as kernel args)
// ---------------------------------------------------------------------------
struct Ptr5 { const float* p[5]; };
struct MixP {
  const float* w[5];     // per-l3 path weights
  const float* bscalar;  // scalar bias (l3==0)
  const float* k;        // (16, kL) skip gains
  int Lx, Lout, kL, act;
};
struct LinP { const float* W[5]; const float* b; };

// ---------------------------------------------------------------------------
// channel_norm (per entity)
// ---------------------------------------------------------------------------
__global__ void norm_kernel(const float* __restrict__ in, float* __restrict__ out,
                            const float* __restrict__ gain, int L, int nEnt){
  int b = blockIdx.x * blockDim.x + threadIdx.x;
  if (b >= nEnt) return;
  int dim = (L+1)*(L+1);
  const float* xp = in + (long)b * 16 * dim;
  float* op = out + (long)b * 16 * dim;
  for (int l = 0; l <= L; ++l) {
    int d = 2*l+1, base = l*l;
    float mu = 0.f;
    if (l == 0) {
      for (int c = 0; c < 16; ++c) mu += xp[c*dim];
      mu *= (1.f/16.f);
    }
    float var = 0.f;
    for (int c = 0; c < 16; ++c)
      for (int m = 0; m < d; ++m) {
        float v = xp[c*dim + base + m] - ((l == 0) ? mu : 0.f);
        var += v * v;
      }
    var /= (16.f * (float)d);
    float rs = rsqrtf(var + 1e-6f);
    for (int c = 0; c < 16; ++c) {
      float g = gain[c*(L+1) + l] * rs;
      for (int m = 0; m < d; ++m)
        op[c*dim + base + m] = (xp[c*dim + base + m] - ((l == 0) ? mu : 0.f)) * g;
    }
  }
}

// ---------------------------------------------------------------------------
// selfmix (optionally preceded by silu on l=0); one thread per (entity,channel)
// ---------------------------------------------------------------------------
__global__ void selfmix_kernel(const float* __restrict__ in, float* __restrict__ out,
                               const float* __restrict__ cg, MixP p, int nEnt){
  int t = blockIdx.x * blockDim.x + threadIdx.x;
  if (t >= nEnt * 16) return;
  int b = t >> 4, c = t & 15;
  int din = (p.Lx+1)*(p.Lx+1), dout = (p.Lout+1)*(p.Lout+1);
  const float* xp = in + ((long)b * 16 + c) * din;
  float x[25];
  for (int i = 0; i < din; ++i) x[i] = xp[i];
  if (p.act) { float v = x[0]; x[0] = v / (1.f + __expf(-v)); }  // silu on l=0
  float* op = out + ((long)b * 16 + c) * dout;
  for (int l3 = 0; l3 <= p.Lout; ++l3) {
    int d3 = 2*l3+1;
    float o[9];
    for (int m = 0; m < d3; ++m) o[m] = 0.f;
    int pc = 0;
    for (int l1 = 0; l1 <= p.Lx; ++l1)
      for (int l2 = 0; l2 <= p.Lx; ++l2) {
        if (iabs_(l1 - l2) > l3 || l3 > l1 + l2) continue;
        float wv = p.w[l3][pc++];
        const float* C = cg + cg_off(l1, l2, l3);
        int d2 = 2*l2+1, d1 = 2*l1+1;
        for (int i = 0; i < d1; ++i) {
          float xi = x[l1*l1 + i] * wv;
          for (int j = 0; j < d2; ++j) {
            float xij = xi * x[l2*l2 + j];
            const float* Cr = C + (i*d2 + j) * d3;
            for (int m = 0; m < d3; ++m) o[m] += Cr[m] * xij;
          }
        }
      }
    float scale = sqrtf((float)d3) / sqrtf((float)pc);
    float bb = (l3 == 0) ? p.bscalar[0] : 0.f;
    float kc = (l3 <= p.Lx) ? p.k[c * p.kL + l3] : 0.f;
    for (int m = 0; m < d3; ++m) {
      float v = o[m] * scale + bb;
      if (l3 <= p.Lx) v += kc * x[l3*l3 + m];
      op[l3*l3 + m] = v;
    }
  }
}

// ---------------------------------------------------------------------------
// channel_mix via V_WMMA_F32_16X16X4_F32: out[b,d,m] = 0.25*sum_c in[b,c,m]*W[c,d]
// One wave per 16-column tile; columns = (entity, m) pairs for irrep l=blockIdx.y.
// REQUIRES nEnt % 16 == 0 (always 4096/16384 here) -> every tile is full, so
// the B loads are unconditional (no exec-mask churn between WMMAs).
// ---------------------------------------------------------------------------
__global__ void channel_mix_kernel(const float* __restrict__ in, float* __restrict__ out,
                                   const float* __restrict__ res, Ptr5 W,
                                   int nEnt, int L, int resDim){
  int l = blockIdx.y;
  int dim = (L+1)*(L+1);
  int tl = 2*l+1;
  long ncols = (long)nEnt * tl;
  int wave = threadIdx.x >> 5;
  long tile = (long)blockIdx.x * (blockDim.x >> 5) + wave;
  if (tile * 16 >= ncols) return;            // uniform per wave
  int lane = threadIdx.x & 31;
  int half = lane >> 4, lr = lane & 15;
  const float* Wl = W.p[l];
  long q = tile * 16 + lr;                   // always in range (nEnt%16==0)
  int ent = (int)(q / tl), m = (int)(q % tl);
  const float* src = in + ((long)ent * 16) * dim + l*l + m;   // + c*dim
  v8f acc = {};
#pragma unroll
  for (int k0 = 0; k0 < 16; k0 += 4) {
    int ka = k0 + 2*half;
    v2f a, bv;
    // A = W^T chunk: A[d=lr][c=ka..ka+1]  (A rows in lanes, K pair in 2 VGPRs)
    a.x = Wl[(ka+0)*16 + lr];
    a.y = Wl[(ka+1)*16 + lr];
    // B = activation chunk: B[c][col]
    bv.x = src[(ka+0)*dim];
    bv.y = src[(ka+1)*dim];
    acc = __builtin_amdgcn_wmma_f32_16x16x4_f32(false, a, false, bv, (short)0, acc,
                                                false, false);
  }
  float* dst = out + ((long)ent * 16) * dim + l*l + m;
  bool doRes = (res != nullptr) && (l*l < resDim);
  const float* rp = doRes ? (res + ((long)ent * 16) * resDim + l*l + m) : nullptr;
#pragma unroll
  for (int r = 0; r < 8; ++r) {
    int d = r + 8*half;                       // C/D: VGPR r = row r / r+8 per lane half
    float v = acc[r] * 0.25f;                 // 1/sqrt(16)
    if (doRes) v += rp[d * resDim];
    dst[d * dim] = v;
  }
}

// ---------------------------------------------------------------------------
// pairmix weight GEMM via WMMA f32: out[e,j] = dist[e,:]@W[:,j] + b[j]
// M=16384, K=32, N=240 (15 tiles of 16)
// ---------------------------------------------------------------------------
__global__ void pairw_gemm_kernel(const float* __restrict__ A, const float* __restrict__ Wm,
                                  const float* __restrict__ bias, float* __restrict__ out,
                                  int M){
  const int N = 240, K = 32;
  int wave = threadIdx.x >> 5;
  int tile = blockIdx.x * (blockDim.x >> 5) + wave;
  int tm = tile / (N/16), tn = tile % (N/16);
  if (tm * 16 >= M) return;
  int lane = threadIdx.x & 31, half = lane >> 4, lr = lane & 15;
  v8f acc = {};
  const float* Arow = A + (long)(tm*16 + lr) * K;
#pragma unroll
  for (int k0 = 0; k0 < K; k0 += 4) {
    int ka = k0 + 2*half;
    v2f a, bv;
    a.x = Arow[ka+0];
    a.y = Arow[ka+1];
    bv.x = Wm[(ka+0)*N + tn*16 + lr];
    bv.y = Wm[(ka+1)*N + tn*16 + lr];
    acc = __builtin_amdgcn_wmma_f32_16x16x4_f32(false, a, false, bv, (short)0, acc,
                                                false, false);
  }
  int col = tn*16 + lr;
  float bb = bias[col];
#pragma unroll
  for (int r = 0; r < 8; ++r) {
    int row = tm*16 + r + 8*half;
    out[(long)row * N + col] = acc[r] + bb;
  }
}

// ---------------------------------------------------------------------------
// pairmix FCTP: per (edge, channel), 15 paths with per-edge weights
// ---------------------------------------------------------------------------
__global__ void pair_fctp_kernel(const float* __restrict__ cf, const float* __restrict__ nf,
                                 const int* __restrict__ edges, const float* __restrict__ pw,
                                 float* __restrict__ out, const float* __restrict__ cg, int nE){
  int t = blockIdx.x * blockDim.x + threadIdx.x;
  if (t >= nE * 16) return;
  int e = t >> 4, c = t & 15;
  int ic = edges[e*2 + 1];   // center = edge_ind[:,1]
  int in_ = edges[e*2 + 0];  // neighbor = edge_ind[:,0]
  const float* xr = cf + ((long)ic * 16 + c) * 9;
  const float* yr = nf + ((long)in_ * 16 + c) * 9;
  float x[9], y[9], o[9];
  for (int i = 0; i < 9; ++i) { x[i] = xr[i]; y[i] = yr[i]; o[i] = 0.f; }
  const float* wp = pw + (long)e * 240 + c * 15;
  const int P1[15] = {0,1,2, 0,1,1,1,2,2, 0,1,1,2,2,2};
  const int P2[15] = {0,1,2, 1,0,1,2,1,2, 2,1,2,0,1,2};
  const int P3[15] = {0,0,0, 1,1,1,1,1,1, 2,2,2,2,2,2};
#pragma unroll
  for (int p = 0; p < 15; ++p) {
    int l1 = P1[p], l2 = P2[p], l3 = P3[p];
    float wv = wp[p];
    const float* C = cg + cg_off_ce(P1[p], P2[p], P3[p]);   // compile-time offset
    int d1 = 2*l1+1, d2 = 2*l2+1, d3 = 2*l3+1;
    for (int i = 0; i < d1; ++i) {
      float xi = x[l1*l1 + i] * wv;
      for (int j = 0; j < d2; ++j) {
        float xij = xi * y[l2*l2 + j];
        const float* Cr = C + (i*d2 + j) * d3;
        for (int m = 0; m < d3; ++m) o[l3*l3 + m] += Cr[m] * xij;
      }
    }
  }
  const float S3[3] = {0.57735027f, 0.70710678f, 0.91287093f}; // sqrt(2l+1)/sqrt(P_l)
  float* op = out + ((long)e * 16 + c) * 9;
  for (int l3 = 0; l3 < 3; ++l3)
    for (int m = 0; m < 2*l3+1; ++m)
      op[l3*l3 + m] = o[l3*l3 + m] * S3[l3];
}

// ---------------------------------------------------------------------------
// lin_to_pair_irreps via WMMA: feats_l[b,k,m] = 0.25*sum_c in[b,c,l^2+m]*W_l[c,k]
// Same tiling as channel_mix; D rows k >= MUL[l] come from clamped (garbage)
// A rows and are simply not stored (WMMA rows are independent).
// Output layout per entity: [l=0:6x1 | l=1:8x3 | l=2:8x5 | l=3:3x7 | l=4:1x9]
// ---------------------------------------------------------------------------
__global__ void lin_wmma_kernel(const float* __restrict__ in, float* __restrict__ out,
                                LinP p, int nEnt){
  const int MUL[5]  = {6, 8, 8, 3, 1};
  const int LOFF[5] = {0, 6, 30, 70, 91};
  int l = blockIdx.y;
  int tl = 2*l+1;
  int K = MUL[l];
  long ncols = (long)nEnt * tl;
  int wave = threadIdx.x >> 5;
  long tile = (long)blockIdx.x * (blockDim.x >> 5) + wave;
  if (tile * 16 >= ncols) return;
  int lane = threadIdx.x & 31;
  int half = lane >> 4, lr = lane & 15;
  const float* Wl = p.W[l];
  int wmax = 16*K - 1;
  long q = tile * 16 + lr;
  int ent = (int)(q / tl), m = (int)(q % tl);
  const float* src = in + ((long)ent * 16) * 25 + l*l + m;    // + c*25
  v8f acc = {};
#pragma unroll
  for (int k0 = 0; k0 < 16; k0 += 4) {
    int ka = k0 + 2*half;
    v2f a, bv;
    int i0 = (ka+0)*K + lr; if (i0 > wmax) i0 = wmax;   // clamped, always in-bounds
    int i1 = (ka+1)*K + lr; if (i1 > wmax) i1 = wmax;
    a.x = Wl[i0];
    a.y = Wl[i1];
    bv.x = src[(ka+0)*25];
    bv.y = src[(ka+1)*25];
    acc = __builtin_amdgcn_wmma_f32_16x16x4_f32(false, a, false, bv, (short)0, acc,
                                                false, false);
  }
  float* dst = out + (long)ent * 100 + LOFF[l] + m;
#pragma unroll
  for (int r = 0; r < 8; ++r) {
    int k = r + 8*half;
    if (k < K) {
      float v = acc[r] * 0.25f;               // 1/sqrt(16)
      if (l == 0) v += p.b[k];
      dst[k * tl] = v;
    }
  }
}

// ---------------------------------------------------------------------------
// block_mapping: per (entity, 4x4 row/col block pair)
// ---------------------------------------------------------------------------
__global__ void blockmap_kernel(const float* __restrict__ feats, float* __restrict__ out,
                                const float* __restrict__ cg, int nEnt){
  int t = blockIdx.x * blockDim.x + threadIdx.x;
  if (t >= nEnt * 16) return;
  int b = t >> 4, pr = t & 15;
  int ri = pr >> 2, ci = pr & 3;
  const int RL[4]   = {0, 0, 1, 2};
  const int RO[4]   = {0, 1, 2, 5};
  const int LOFF[5] = {0, 6, 30, 70, 91};
  int lr = RL[ri], lc = RL[ci];
  int dr = 2*lr+1, dc = 2*lc+1;
  // deterministic counter start (consumption order over prior blocks)
  int cnt[5] = {0,0,0,0,0};
  for (int q = 0; q < pr; ++q) {
    int qlr = RL[q >> 2], qlc = RL[q & 3];
    for (int l3 = iabs_(qlr - qlc); l3 <= qlr + qlc; ++l3) cnt[l3]++;
  }
  float blk[25];
  for (int z = 0; z < dr*dc; ++z) blk[z] = 0.f;
  for (int l3 = iabs_(lr - lc); l3 <= lr + lc; ++l3) {
    int d3 = 2*l3+1;
    const float* xs = feats + (long)b * 100 + LOFF[l3] + cnt[l3] * d3;
    cnt[l3]++;
    const float* C = cg + cg_off(lr, lc, l3);
    for (int i = 0; i < dr; ++i)
      for (int j = 0; j < dc; ++j) {
        const float* Cr = C + (i*dc + j) * d3;
        float a = 0.f;
        for (int m = 0; m < d3; ++m) a += Cr[m] * xs[m];
        blk[i*dc + j] += a;
      }
  }
  int r0 = RO[ri], c0 = RO[ci];
  for (int i = 0; i < dr; ++i)
    for (int j = 0; j < dc; ++j)
      out[(long)b * 100 + (r0 + i) * 10 + (c0 + j)] = blk[i*dc + j];
}

// ===========================================================================
// Host launch
// ===========================================================================
struct SphP { Ptr5 cm; MixP mix; };
struct BlockParams { const float* gain; SphP s1, s2; };

extern "C" void kernel_launch(void* const* d_in, const int* in_sizes, int n_in,
                              void* d_out, int out_size, void* d_ws, size_t ws_size,
                              hipStream_t stream) {
  (void)in_sizes; (void)n_in; (void)out_size; (void)ws_size;
  const float* nodes = (const float*)d_in[0];     // (4096,16,9)
  const int*   edges = (const int*)d_in[1];       // (16384,2)
  /* d_in[2] = Y, unused by the reference forward */
  const float* dist  = (const float*)d_in[3];     // (16384,32)

  // ---- unpack params (jax pytree order: dicts sorted by key, lists in order)
  int pi = 4;
  auto nxt = [&]() { return (const float*)d_in[pi++]; };
  auto read_sph = [&](int Lx, int Lout) {
    SphP s{};
    for (int l = 0; l <= Lout; ++l) s.cm.p[l] = nxt();
    for (int l = Lout + 1; l < 5; ++l) s.cm.p[l] = nullptr;
    s.mix.bscalar = nxt();
    s.mix.k = nxt();
    for (int l = 0; l <= Lout; ++l) s.mix.w[l] = nxt();
    for (int l = Lout + 1; l < 5; ++l) s.mix.w[l] = nullptr;
    s.mix.Lx = Lx; s.mix.Lout = Lout;
    s.mix.kL = ((Lx < Lout) ? Lx : Lout) + 1;
    s.mix.act = 1;
    return s;
  };
  auto read_block = [&](int Lx, int L) {
    BlockParams b{};
    b.gain = nxt();
    b.s1 = read_sph(Lx, L);
    b.s2 = read_sph(L, L);
    return b;
  };

  const float* diag_gain = nxt();                               // diag_norm_gain
  LinP lin_diag{}; for (int l = 0; l < 5; ++l) lin_diag.W[l] = nxt(); lin_diag.b = nxt();
  LinP lin_off{};  for (int l = 0; l < 5; ++l) lin_off.W[l]  = nxt(); lin_off.b  = nxt();
  const float* off_gain = nxt();                                // offdiag_norm_gain
  const float* pmW = nxt(); const float* pmB = nxt();           // pairmix W,b
  BlockParams res_center   = read_block(2, 2);
  BlockParams res_diag     = read_block(2, 4);
  BlockParams res_neighbor = read_block(2, 2);
  BlockParams res_offdiag  = read_block(2, 4);

  // ---- workspace carve
  char* wsb = (char*)d_ws;
  auto carve = [&](size_t bytes) {
    void* p = (void*)wsb;
    wsb += (bytes + 255) & ~(size_t)255;
    return p;
  };
  float* CGV = (float*)carve(13104 * sizeof(float));
  const size_t BIG = (size_t)16384 * 400 * sizeof(float);
  float* B1 = (float*)carve(BIG);
  float* B2 = (float*)carve(BIG);
  float* B3 = (float*)carve(BIG);
  float* CF = (float*)carve((size_t)4096 * 144 * sizeof(float));
  float* NF = (float*)carve((size_t)4096 * 144 * sizeof(float));
  float* D4 = (float*)carve((size_t)4096 * 400 * sizeof(float));

  auto gsz = [](long n, int b) { return (int)((n + b - 1) / b); };

  // ---- CG table (65 valid (l1,l2,l3) triples)
  cg_init_kernel<<<65, 64, 0, stream>>>(CGV);

  // ---- residual block driver
  auto run_block = [&](const float* in, float* outp, int nEnt, const BlockParams& bp,
                       int Lx, int L, float* t0, float* t1, float* t2) {
    int dimx = (Lx+1)*(Lx+1);
    norm_kernel<<<gsz(nEnt,128),128,0,stream>>>(in, t0, bp.gain, Lx, nEnt);
    selfmix_kernel<<<gsz((long)nEnt*16,128),128,0,stream>>>(t0, t1, CGV, bp.s1.mix, nEnt);
    long ncolsMax = (long)nEnt * (2*L+1);
    dim3 gmix((unsigned)gsz((ncolsMax + 15)/16, 8), (unsigned)(L+1));
    channel_mix_kernel<<<gmix,256,0,stream>>>(t1, t2, nullptr, bp.s1.cm, nEnt, L, 0);
    selfmix_kernel<<<gsz((long)nEnt*16,128),128,0,stream>>>(t2, t1, CGV, bp.s2.mix, nEnt);
    channel_mix_kernel<<<gmix,256,0,stream>>>(t1, outp, in, bp.s2.cm, nEnt, L, dimx);
  };

  // ---- node-side blocks
  run_block(nodes, CF, 4096, res_center,   2, 2, B1, B2, B3);
  run_block(nodes, NF, 4096, res_neighbor, 2, 2, B1, B2, B3);
  run_block(CF,    D4, 4096, res_diag,     2, 4, B1, B2, B3);

  // ---- pairmix: weight GEMM (WMMA) + gather/FCTP
  pairw_gemm_kernel<<<(16384/16)*(240/16)/8, 256, 0, stream>>>(dist, pmW, pmB, B1, 16384);
  pair_fctp_kernel<<<gsz((long)16384*16,128),128,0,stream>>>(CF, NF, edges, B1, B3, CGV, 16384);

  // ---- offdiag residual block: in=B3, out=B2 (t0/t2 alias B2, t1=B1)
  run_block(B3, B2, 16384, res_offdiag, 2, 4, B2, B1, B2);

  float* outF = (float*)d_out;

  // ---- diag finalize: norm -> lin (WMMA) -> block mapping
  norm_kernel<<<gsz(4096,128),128,0,stream>>>(D4, B1, diag_gain, 4, 4096);
  {
    dim3 g((unsigned)gsz(((long)4096*9 + 15)/16, 8), 5u);
    lin_wmma_kernel<<<g,256,0,stream>>>(B1, B3, lin_diag, 4096);
  }
  blockmap_kernel<<<gsz((long)4096*16,128),128,0,stream>>>(B3, outF, CGV, 4096);

  // ---- offdiag finalize
  norm_kernel<<<gsz(16384,128),128,0,stream>>>(B2, B1, off_gain, 4, 16384);
  {
    dim3 g((unsigned)gsz(((long)16384*9 + 15)/16, 8), 5u);
    lin_wmma_kernel<<<g,256,0,stream>>>(B1, B3, lin_off, 16384);
  }
  blockmap_kernel<<<gsz((long)16384*16,128),128,0,stream>>>(B3, outF + (long)4096*100, CGV, 16384);
}